// SynthesisNetwork_23742579212920
// MI455X (gfx1250) — compile-verified
//
#include <hip/hip_runtime.h>
#include <math.h>

#define B_ 8192
#define H_ 400
#define A_ 73
#define U_ 64
#define K_ 10
#define M_ 20

typedef float v2f __attribute__((ext_vector_type(2)));
typedef float v8f __attribute__((ext_vector_type(8)));

__device__ __forceinline__ float sigm(float x) { return 1.0f / (1.0f + expf(-x)); }

// ---------------------------------------------------------------------------
// GEMM: C[M x NDIM] = A[M x K] @ W[K x NDIM] + bias[NDIM]
// NDIM is a compile-time template parameter so every B-load offset (k*NDIM*4)
// folds into the instruction's 24-bit immediate offset: the unrolled loop body
// becomes pure {global_load_b64(A), 16x global_load_b32(B, off:imm), 8x WMMA}
// with a single base-pointer bump per iteration.
//
// One wave computes a 16x32 tile (two 16x16 WMMA accumulators sharing the A
// fragment) with V_WMMA_F32_16X16X4_F32, K step = 4.
// Layouts per CDNA5 ISA 7.12.2:
//   A 16x4 f32 : lane l -> m = l&15, kpair = (l>>4)*2 ; v2f = {A[m][k+kp], A[m][k+kp+1]}
//   B 4x16 f32 : lane l -> n = l&15, kpair = (l>>4)*2 ; v2f = {W[k+kp][n], W[k+kp+1][n]}
//   C/D 16x16  : vgpr r, lane l -> row = r + 8*(l>>4), col = l&15
// Out-of-range N columns: load address is CLAMPED (no branches in the loop) —
// D[m][n] depends only on B column n, and those columns are masked at store.
// Requires Kdim % 4 == 0 (476, 876, 400, 1200) and Mdim % 16 == 0 (B_ = 8192).
// ---------------------------------------------------------------------------
#define GEMM_WAVES 4
template <int NDIM>
__global__ __launch_bounds__(32 * GEMM_WAVES)
void gemm_wmma_f32(const float* __restrict__ Amat, const float* __restrict__ Wmat,
                   const float* __restrict__ bias, float* __restrict__ Cmat,
                   int Mdim, int Kdim)
{
    const int lane = threadIdx.x & 31;
    const int wid  = threadIdx.x >> 5;
    const int tn = blockIdx.x * 32;
    const int tm = (blockIdx.y * GEMM_WAVES + wid) * 16;
    if (tm >= Mdim) return;

    const int ma = lane & 15;
    const int kp = (lane >> 4) << 1;      // 0 or 2
    const int nb = lane & 15;

    const int n0 = tn + nb;
    const int n1 = tn + 16 + nb;
    const int n0c = (n0 < NDIM) ? n0 : (NDIM - 1);   // clamped load column
    const int n1c = (n1 < NDIM) ? n1 : (NDIM - 1);

    const float* __restrict__ arow = Amat + (size_t)(tm + ma) * Kdim + kp;
    const float* __restrict__ w0   = Wmat + (size_t)kp * NDIM + n0c;
    const float* __restrict__ w1   = Wmat + (size_t)kp * NDIM + n1c;

    v8f acc0 = {0.f, 0.f, 0.f, 0.f, 0.f, 0.f, 0.f, 0.f};
    v8f acc1 = {0.f, 0.f, 0.f, 0.f, 0.f, 0.f, 0.f, 0.f};

#pragma unroll 4
    for (int k = 0; k < Kdim; k += 4) {
        // A fragment: one aligned 8B load per lane (Kdim, kp even -> 8B aligned)
        const v2f a = *(const v2f*)(arow + k);
        const int o = k * NDIM;            // compile-time stride -> imm offsets
        v2f b0, b1;
        b0.x = w0[o];
        b0.y = w0[o + NDIM];
        b1.x = w1[o];
        b1.y = w1[o + NDIM];
        acc0 = __builtin_amdgcn_wmma_f32_16x16x4_f32(false, a, false, b0,
                                                     (short)0, acc0, false, false);
        acc1 = __builtin_amdgcn_wmma_f32_16x16x4_f32(false, a, false, b1,
                                                     (short)0, acc1, false, false);
    }

    const int row0 = tm + ((lane >> 4) << 3);
    if (n0 < NDIM) {
        const float bs = bias ? bias[n0] : 0.f;
#pragma unroll
        for (int r = 0; r < 8; ++r)
            Cmat[(size_t)(row0 + r) * NDIM + n0] = acc0[r] + bs;
    }
    if (n1 < NDIM) {
        const float bs = bias ? bias[n1] : 0.f;
#pragma unroll
        for (int r = 0; r < 8; ++r)
            Cmat[(size_t)(row0 + r) * NDIM + n1] = acc1[r] + bs;
    }
}

// ---------------------------------------------------------------------------
// Row-wise concat of up to 4 matrices: dst[B x (c0+c1+c2+c3)]
// ---------------------------------------------------------------------------
__global__ void build_cat4(float* __restrict__ dst,
                           const float* __restrict__ p0, int c0,
                           const float* __restrict__ p1, int c1,
                           const float* __restrict__ p2, int c2,
                           const float* __restrict__ p3, int c3)
{
    const int ld = c0 + c1 + c2 + c3;
    size_t idx = (size_t)blockIdx.x * blockDim.x + threadIdx.x;
    size_t total = (size_t)B_ * ld;
    if (idx >= total) return;
    int row = (int)(idx / ld);
    int col = (int)(idx % ld);
    float v;
    if (col < c0)                v = p0[(size_t)row * c0 + col];
    else if (col < c0 + c1)      v = p1[(size_t)row * c1 + (col - c0)];
    else if (col < c0 + c1 + c2) v = p2[(size_t)row * c2 + (col - c0 - c1)];
    else                         v = p3[(size_t)row * c3 + (col - c0 - c1 - c2)];
    dst[idx] = v;
}

// ---------------------------------------------------------------------------
// Peephole LSTM cell elementwise: z[B x 4H] (bias already fused in GEMM)
// ---------------------------------------------------------------------------
__global__ void lstm_cell(const float* __restrict__ z, const float* __restrict__ cprev,
                          const float* __restrict__ p,
                          float* __restrict__ hout, float* __restrict__ cout)
{
    size_t idx = (size_t)blockIdx.x * blockDim.x + threadIdx.x;
    if (idx >= (size_t)B_ * H_) return;
    int b = (int)(idx / H_);
    int j = (int)(idx % H_);
    const float* zr = z + (size_t)b * (4 * H_);
    float c0 = cprev[idx];
    float i  = sigm(zr[j]          + p[j]          * c0);
    float f  = sigm(zr[H_ + j]     + p[H_ + j]     * c0);
    float g  = tanhf(zr[2 * H_ + j]);
    float cn = f * c0 + i * g;
    float o  = sigm(zr[3 * H_ + j] + p[2 * H_ + j] * cn);
    hout[idx] = o * tanhf(cn);
    cout[idx] = cn;
}

// ---------------------------------------------------------------------------
// Gaussian attention window: one block (128 threads) per batch row
// att[B x 3K] -> kn[B x K], phi[B x U], wn[B x A]
// ---------------------------------------------------------------------------
__global__ __launch_bounds__(128)
void attention_k(const float* __restrict__ att, const float* __restrict__ kin,
                 const float* __restrict__ cseq,
                 float* __restrict__ kn_out, float* __restrict__ phi_out,
                 float* __restrict__ wn_out)
{
    __shared__ float s_al[K_], s_be[K_], s_kn[K_], s_phi[U_];
    const int b = blockIdx.x;
    const int t = threadIdx.x;
    if (t < K_) {
        const float* ar = att + (size_t)b * (3 * K_);
        float al = expf(ar[t]);
        float be = expf(ar[K_ + t]);
        float ka = expf(ar[2 * K_ + t]);
        float kn = kin[(size_t)b * K_ + t] + ka;
        s_al[t] = al; s_be[t] = be; s_kn[t] = kn;
        kn_out[(size_t)b * K_ + t] = kn;
    }
    __syncthreads();
    if (t < U_) {
        float u = (float)t, s = 0.f;
#pragma unroll
        for (int kk = 0; kk < K_; ++kk) {
            float d = s_kn[kk] - u;
            s += s_al[kk] * expf(-s_be[kk] * d * d);
        }
        s_phi[t] = s;
        phi_out[(size_t)b * U_ + t] = s;
    }
    __syncthreads();
    if (t < A_) {
        const float* cs = cseq + (size_t)b * U_ * A_ + t;
        float s = 0.f;
        for (int u = 0; u < U_; ++u) s += s_phi[u] * cs[(size_t)u * A_];
        wn_out[(size_t)b * A_ + t] = s;
    }
}

// ---------------------------------------------------------------------------
// MDN head: o[B x 121] -> eos / pi(softmax) / mu / sigma(exp) / rho(tanh)
// ---------------------------------------------------------------------------
__global__ void mdn_out(const float* __restrict__ o, float* __restrict__ out)
{
    int b = blockIdx.x * blockDim.x + threadIdx.x;
    if (b >= B_) return;
    const float* orow = o + (size_t)b * 121;
    out[b] = sigm(orow[0]);                                   // eos
    float mx = orow[1];
#pragma unroll
    for (int j = 1; j < M_; ++j) mx = fmaxf(mx, orow[1 + j]);
    float e[M_]; float sum = 0.f;
#pragma unroll
    for (int j = 0; j < M_; ++j) { e[j] = expf(orow[1 + j] - mx); sum += e[j]; }
    float inv = 1.f / sum;
    float* pi  = out + (size_t)B_ * 1;
    float* mu1 = out + (size_t)B_ * 21;
    float* mu2 = out + (size_t)B_ * 41;
    float* s1  = out + (size_t)B_ * 61;
    float* s2  = out + (size_t)B_ * 81;
    float* rho = out + (size_t)B_ * 101;
#pragma unroll
    for (int j = 0; j < M_; ++j) {
        size_t o2 = (size_t)b * M_ + j;
        pi[o2]  = e[j] * inv;
        mu1[o2] = orow[21 + j];
        mu2[o2] = orow[41 + j];
        s1[o2]  = expf(orow[61 + j]);
        s2[o2]  = expf(orow[81 + j]);
        rho[o2] = tanhf(orow[101 + j]);
    }
}

// ---------------------------------------------------------------------------
extern "C" void kernel_launch(void* const* d_in, const int* in_sizes, int n_in,
                              void* d_out_v, int out_size, void* d_ws, size_t ws_size,
                              hipStream_t stream)
{
    (void)in_sizes; (void)n_in; (void)out_size; (void)ws_size;

    const float* x    = (const float*)d_in[0];
    const float* cseq = (const float*)d_in[1];
    const float* h1   = (const float*)d_in[2];
    const float* c1   = (const float*)d_in[3];
    const float* h2   = (const float*)d_in[4];
    const float* c2   = (const float*)d_in[5];
    const float* h3   = (const float*)d_in[6];
    const float* c3   = (const float*)d_in[7];
    const float* kin  = (const float*)d_in[8];
    const float* w    = (const float*)d_in[9];
    const float* Wx1  = (const float*)d_in[10]; const float* Wh1 = (const float*)d_in[11];
    const float* p1   = (const float*)d_in[12]; const float* b1  = (const float*)d_in[13];
    const float* Wx2  = (const float*)d_in[14]; const float* Wh2 = (const float*)d_in[15];
    const float* p2   = (const float*)d_in[16]; const float* b2  = (const float*)d_in[17];
    const float* Wx3  = (const float*)d_in[18]; const float* Wh3 = (const float*)d_in[19];
    const float* p3   = (const float*)d_in[20]; const float* b3  = (const float*)d_in[21];
    const float* Wa   = (const float*)d_in[22]; const float* ba  = (const float*)d_in[23];
    const float* Wm   = (const float*)d_in[24]; const float* bm  = (const float*)d_in[25];
    const float* Wc   = (const float*)d_in[26]; const float* bc  = (const float*)d_in[27];

    float* out = (float*)d_out_v;
    float* ws  = (float*)d_ws;

    // workspace layout (floats)
    float* wcat1 = ws;                                    // 476*1600
    float* wcat2 = wcat1 + (size_t)476 * 1600;            // 876*1600
    float* wcat3 = wcat2 + (size_t)876 * 1600;            // 876*1600
    float* xcat  = wcat3 + (size_t)876 * 1600;            // B*1200 (max of 476/876/1200)
    float* zbuf  = xcat  + (size_t)B_ * 1200;             // B*1600
    float* attb  = zbuf;                                  // alias (B*30, after z1 consumed)
    float* obuf  = zbuf;                                  // alias (B*121, after z3 consumed)

    // d_out slices (reference return order)
    float* clo  = out + (size_t)B_ * 121;   // char_logits
    float* h1n  = out + (size_t)B_ * 194;
    float* c1n  = out + (size_t)B_ * 594;
    float* h2n  = out + (size_t)B_ * 994;
    float* c2n  = out + (size_t)B_ * 1394;
    float* h3n  = out + (size_t)B_ * 1794;
    float* c3n  = out + (size_t)B_ * 2194;
    float* kno  = out + (size_t)B_ * 2594;
    float* wno  = out + (size_t)B_ * 2604;
    float* phio = out + (size_t)B_ * 2677;

    // Row-concat weights [Wx; Wh] are contiguous -> plain d2d copies
    hipMemcpyAsync(wcat1,                    Wx1, sizeof(float) * 76  * 1600, hipMemcpyDeviceToDevice, stream);
    hipMemcpyAsync(wcat1 + (size_t)76*1600,  Wh1, sizeof(float) * 400 * 1600, hipMemcpyDeviceToDevice, stream);
    hipMemcpyAsync(wcat2,                    Wx2, sizeof(float) * 476 * 1600, hipMemcpyDeviceToDevice, stream);
    hipMemcpyAsync(wcat2 + (size_t)476*1600, Wh2, sizeof(float) * 400 * 1600, hipMemcpyDeviceToDevice, stream);
    hipMemcpyAsync(wcat3,                    Wx3, sizeof(float) * 476 * 1600, hipMemcpyDeviceToDevice, stream);
    hipMemcpyAsync(wcat3 + (size_t)476*1600, Wh3, sizeof(float) * 400 * 1600, hipMemcpyDeviceToDevice, stream);

    const int blk = 256;
    const dim3 gz(1600 / 32, B_ / (16 * GEMM_WAVES));   // N tiles of 32
    const int gthr = 32 * GEMM_WAVES;
    const unsigned ew = (unsigned)(((size_t)B_ * H_ + blk - 1) / blk);

    // ---- LSTM1 ----
    build_cat4<<<(unsigned)(((size_t)B_ * 476 + blk - 1) / blk), blk, 0, stream>>>(
        xcat, x, 3, w, 73, h1, 400, nullptr, 0);
    gemm_wmma_f32<1600><<<gz, gthr, 0, stream>>>(xcat, wcat1, b1, zbuf, B_, 476);
    lstm_cell<<<ew, blk, 0, stream>>>(zbuf, c1, p1, h1n, c1n);

    // ---- Attention ----
    const dim3 ga(1, B_ / (16 * GEMM_WAVES));           // N=30 -> one 32-col tile
    gemm_wmma_f32<30><<<ga, gthr, 0, stream>>>(h1n, Wa, ba, attb, B_, 400);
    attention_k<<<B_, 128, 0, stream>>>(attb, kin, cseq, kno, phio, wno);

    // ---- LSTM2 ----
    build_cat4<<<(unsigned)(((size_t)B_ * 876 + blk - 1) / blk), blk, 0, stream>>>(
        xcat, x, 3, h1n, 400, wno, 73, h2, 400);
    gemm_wmma_f32<1600><<<gz, gthr, 0, stream>>>(xcat, wcat2, b2, zbuf, B_, 876);
    lstm_cell<<<ew, blk, 0, stream>>>(zbuf, c2, p2, h2n, c2n);

    // ---- LSTM3 ----
    build_cat4<<<(unsigned)(((size_t)B_ * 876 + blk - 1) / blk), blk, 0, stream>>>(
        xcat, x, 3, h2n, 400, wno, 73, h3, 400);
    gemm_wmma_f32<1600><<<gz, gthr, 0, stream>>>(xcat, wcat3, b3, zbuf, B_, 876);
    lstm_cell<<<ew, blk, 0, stream>>>(zbuf, c3, p3, h3n, c3n);

    // ---- Output heads ----
    build_cat4<<<(unsigned)(((size_t)B_ * 1200 + blk - 1) / blk), blk, 0, stream>>>(
        xcat, h1n, 400, h2n, 400, h3n, 400, nullptr, 0);
    const dim3 go((121 + 31) / 32, B_ / (16 * GEMM_WAVES));
    gemm_wmma_f32<121><<<go, gthr, 0, stream>>>(xcat, Wm, bm, obuf, B_, 1200);
    const dim3 gc((73 + 31) / 32, B_ / (16 * GEMM_WAVES));
    gemm_wmma_f32<73><<<gc, gthr, 0, stream>>>(xcat, Wc, bc, clo, B_, 1200);
    mdn_out<<<(B_ + blk - 1) / blk, blk, 0, stream>>>(obuf, out);
}